// TFLoudnessLoss_46789373723048
// MI455X (gfx1250) — compile-verified
//
#include <hip/hip_runtime.h>
#include <math.h>

#ifndef M_PI
#define M_PI 3.14159265358979323846
#endif

// ---- static problem config (matches reference) ----
#define NBANDS   8
#define BATCH    32
#define T_IN     160000
#define WIN      2048
#define STRIDEW  1024
#define NSEG     155            // (160000-2048)/1024 + 1
#define CHUNK    1024
#define NCHUNK   156            // covers t < 159744 = all samples any segment touches
#define T_USE    (NCHUNK*CHUNK) // 159744
#define KTAPS    128            // truncated IR of weighting biquad (poles 0.9 -> |h[128]|~2e-4)
#define NTILE    9              // Toeplitz offsets d = 0..128 step 16
#define SEG_GROUPS 10           // ceil(155/16)

typedef __attribute__((ext_vector_type(2))) float v2f;
typedef __attribute__((ext_vector_type(8))) float v8f;

struct C10d { double v[10]; };  // hb0,hb1,hb2,ha1,ha2, lb0,lb1,lb2,la1,la2
struct C10f { float  v[10]; };
struct AllC { double v[NBANDS][10]; };

// ---------------------------------------------------------------------------
// Init: per-band 6x6 chunk transition matrix M (6 homogeneous basis runs of
// 1024 steps, f64) + the 128-tap weighting impulse response h.
// ---------------------------------------------------------------------------
__global__ void tfl_init_kernel(AllC cf, double* __restrict__ Mws, float* __restrict__ hws) {
    int tid = threadIdx.x;
    if (tid < NBANDS * 6) {
        int band = tid / 6, j = tid % 6;
        const double* c = cf.v[band];
        double u1 = 0, u2 = 0, h1 = 0, h2 = 0, l1 = 0, l2 = 0;
        if (j == 0) u1 = 1.0; else if (j == 1) u2 = 1.0;
        else if (j == 2) h1 = 1.0; else if (j == 3) h2 = 1.0;
        else if (j == 4) l1 = 1.0; else l2 = 1.0;
        for (int n = 0; n < CHUNK; ++n) {
            double u  = 0.0;
            double yh = c[0]*u  + c[1]*u1 + c[2]*u2 - c[3]*h1 - c[4]*h2;
            double yl = c[5]*yh + c[6]*h1 + c[7]*h2 - c[8]*l1 - c[9]*l2;
            u2 = u1; u1 = u; h2 = h1; h1 = yh; l2 = l1; l1 = yl;
        }
        double out[6] = {u1, u2, h1, h2, l1, l2};
        for (int i = 0; i < 6; ++i) Mws[band*36 + i*6 + j] = out[i];
    } else if (tid == NBANDS * 6) {
        // weighting biquad b=(1,-2,1), a=(1,-1.8,0.81): impulse response
        double x1 = 0, x2 = 0, y1 = 0, y2 = 0;
        for (int n = 0; n < KTAPS; ++n) {
            double x = (n == 0) ? 1.0 : 0.0;
            double y = x - 2.0*x1 + x2 + 1.8*y1 - 0.81*y2;
            hws[n] = (float)y;
            x2 = x1; x1 = x; y2 = y1; y1 = y;
        }
    }
}

// ---------------------------------------------------------------------------
// Pass 1: zero-state run of each 1024-chunk (f64), emit end state S_zs[6].
// tid = input*(32*156) + row*156 + chunk
// ---------------------------------------------------------------------------
__global__ void tfl_pass1_kernel(const float* __restrict__ sig, const float* __restrict__ wm,
                                 C10d cf, double* __restrict__ Szs) {
    int tid = blockIdx.x * blockDim.x + threadIdx.x;
    if (tid >= 2 * BATCH * NCHUNK) return;
    int chunk = tid % NCHUNK;
    int row   = (tid / NCHUNK) % BATCH;
    int inp   = tid / (NCHUNK * BATCH);
    const double* c = cf.v;
    const float* x = (inp == 0 ? sig : wm) + (size_t)row * T_IN + (size_t)chunk * CHUNK;
    double u1 = 0, u2 = 0, h1 = 0, h2 = 0, l1 = 0, l2 = 0;
    for (int n = 0; n < CHUNK; ++n) {
        if ((n & 63) == 0) __builtin_prefetch(x + n + 256, 0, 0);
        double u  = (double)x[n];
        double yh = c[0]*u  + c[1]*u1 + c[2]*u2 - c[3]*h1 - c[4]*h2;
        double yl = c[5]*yh + c[6]*h1 + c[7]*h2 - c[8]*l1 - c[9]*l2;
        u2 = u1; u1 = u; h2 = h1; h1 = yh; l2 = l1; l1 = yl;
    }
    double* o = Szs + (size_t)tid * 6;
    o[0] = u1; o[1] = u2; o[2] = h1; o[3] = h2; o[4] = l1; o[5] = l2;
}

// ---------------------------------------------------------------------------
// Sequential state scan per (input,row): S[c+1] = M*S[c] + Szs[c]  (f64).
// Emits f32 chunk-initial states for pass 2.
// ---------------------------------------------------------------------------
__global__ void tfl_scan_kernel(const double* __restrict__ Mband,
                                const double* __restrict__ Szs,
                                float* __restrict__ Sstart) {
    int tid = threadIdx.x;
    if (tid >= 2 * BATCH) return;
    double M[36];
#pragma unroll
    for (int i = 0; i < 36; ++i) M[i] = Mband[i];
    double S[6] = {0, 0, 0, 0, 0, 0};
    const double* z = Szs + (size_t)tid * NCHUNK * 6;
    float* o = Sstart + (size_t)tid * NCHUNK * 6;
    for (int cc = 0; cc < NCHUNK; ++cc) {
#pragma unroll
        for (int i = 0; i < 6; ++i) o[cc*6 + i] = (float)S[i];
        double Sn[6];
#pragma unroll
        for (int i = 0; i < 6; ++i) {
            double a = z[cc*6 + i];
#pragma unroll
            for (int j = 0; j < 6; ++j) a += M[i*6 + j] * S[j];
            Sn[i] = a;
        }
#pragma unroll
        for (int i = 0; i < 6; ++i) S[i] = Sn[i];
    }
}

// ---------------------------------------------------------------------------
// Pass 2: re-run each chunk with correct initial state (f32), write filtered y.
// ---------------------------------------------------------------------------
__global__ void tfl_pass2_kernel(const float* __restrict__ sig, const float* __restrict__ wm,
                                 C10f cf, const float* __restrict__ Sstart,
                                 float* __restrict__ yband) {
    int tid = blockIdx.x * blockDim.x + threadIdx.x;
    if (tid >= 2 * BATCH * NCHUNK) return;
    int chunk = tid % NCHUNK;
    int row   = (tid / NCHUNK) % BATCH;
    int inp   = tid / (NCHUNK * BATCH);
    const float* c = cf.v;
    const float* x = (inp == 0 ? sig : wm) + (size_t)row * T_IN + (size_t)chunk * CHUNK;
    const float* s = Sstart + (size_t)tid * 6;
    float u1 = s[0], u2 = s[1], h1 = s[2], h2 = s[3], l1 = s[4], l2 = s[5];
    float* y = yband + ((size_t)(inp * BATCH + row)) * T_USE + (size_t)chunk * CHUNK;
    for (int n = 0; n < CHUNK; ++n) {
        if ((n & 63) == 0) __builtin_prefetch(x + n + 256, 0, 0);
        float u  = x[n];
        float yh = c[0]*u  + c[1]*u1 + c[2]*u2 - c[3]*h1 - c[4]*h2;
        float yl = c[5]*yh + c[6]*h1 + c[7]*h2 - c[8]*l1 - c[9]*l2;
        y[n] = yl;
        u2 = u1; u1 = u; h2 = h1; h1 = yh; l2 = l1; l1 = yl;
    }
}

// ---------------------------------------------------------------------------
// Weighting filter + loudness via WMMA: per-window causal FIR (truncated IR,
// zero-padded at window start == zero-init biquad) as banded-Toeplitz GEMM.
// One wave per (input,row,group-of-16-windows). D(16 out x 16 win) accumulated
// over 9 Toeplitz tiles x 4 K-chunks = 36 V_WMMA_F32_16X16X4_F32 per tile.
// ---------------------------------------------------------------------------
__global__ void __launch_bounds__(32)
tfl_wmma_loud_kernel(const float* __restrict__ yband, const float* __restrict__ hws,
                     float* __restrict__ loud) {
    int b    = blockIdx.x;
    int g    = b % SEG_GROUPS;
    int row  = (b / SEG_GROUPS) % BATCH;
    int inp  = b / (SEG_GROUPS * BATCH);
    int lane = threadIdx.x;
    bool hi  = lane >= 16;
    int i16  = lane & 15;
    int nseg = (g == SEG_GROUPS - 1) ? (NSEG - 16 * (SEG_GROUPS - 1)) : 16;
    int seg  = g * 16 + i16;
    int seg_eff = seg < NSEG ? seg : NSEG - 1;  // clamp addr; invalid lanes never write
    const float* base = yband + ((size_t)(inp * BATCH + row)) * T_USE
                              + (size_t)seg_eff * STRIDEW;

    // A tiles: Toeplitz slices of h.  At[t] <-> offset d = 128 - 16t.
    // 16x4 f32 A layout: vgpr0 = K {0(lo),2(hi)}, vgpr1 = K {1(lo),3(hi)}; M = lane&15.
    v2f At[NTILE][4];
#pragma unroll
    for (int t = 0; t < NTILE; ++t) {
        int d = KTAPS - 16 * t;
#pragma unroll
        for (int kk = 0; kk < 4; ++kk) {
            int kb = 4 * kk + (hi ? 2 : 0);
            int i0 = d + i16 - kb;       // h index for K = kb   (vgpr0)
            int i1 = i0 - 1;             // h index for K = kb+1 (vgpr1)
            v2f a;
            a.x = (i0 >= 0 && i0 < KTAPS) ? hws[i0] : 0.0f;
            a.y = (i1 >= 0 && i1 < KTAPS) ? hws[i1] : 0.0f;
            At[t][kk] = a;
        }
    }

    // B ring: Bt[8] = newest tile (base = p), Bt[t] = base p - 16*(8-t).
    v2f Bt[NTILE][4];
#pragma unroll
    for (int t = 0; t < NTILE; ++t)
#pragma unroll
        for (int kk = 0; kk < 4; ++kk) { v2f z; z.x = 0.0f; z.y = 0.0f; Bt[t][kk] = z; }

    float e = 0.0f;
    for (int m = 0; m < WIN / 16; ++m) {
        int p = m * 16;
#pragma unroll
        for (int t = 0; t < NTILE - 1; ++t)
#pragma unroll
            for (int kk = 0; kk < 4; ++kk) Bt[t][kk] = Bt[t + 1][kk];
        // load new 16(samples) x 16(windows) tile at base p; rows per lane:
        // vgpr0 row = 4kk + (hi?2:0), vgpr1 row = that+1 -> one aligned float2 load
#pragma unroll
        for (int kk = 0; kk < 4; ++kk) {
            int r0 = 4 * kk + (hi ? 2 : 0);
            Bt[NTILE - 1][kk] = *reinterpret_cast<const v2f*>(base + p + r0);
        }
        v8f c = {};
#pragma unroll
        for (int t = 0; t < NTILE; ++t)
#pragma unroll
            for (int kk = 0; kk < 4; ++kk)
                c = __builtin_amdgcn_wmma_f32_16x16x4_f32(
                        false, At[t][kk], false, Bt[t][kk], (short)0, c, false, false);
#pragma unroll
        for (int q = 0; q < 8; ++q) e += c[q] * c[q];
    }
    // lanes 0-15 hold M=0..7 partials, 16-31 hold M=8..15; combine halves (wave32)
    e += __shfl_xor(e, 16, 32);
    if (!hi && i16 < nseg) {
        float mean = e * (1.0f / (float)WIN);
        loud[((size_t)inp * BATCH + row) * NSEG + seg] = 10.0f * log10f(mean + 1e-8f);
    }
}

// ---------------------------------------------------------------------------
// Per-band loss = mean |loud_wm - loud_sig| over 32x155.
// ---------------------------------------------------------------------------
__global__ void tfl_band_reduce_kernel(const float* __restrict__ loud,
                                       float* __restrict__ band_loss, int band) {
    __shared__ float sm[256];
    float s = 0.0f;
    for (int idx = threadIdx.x; idx < BATCH * NSEG; idx += 256)
        s += fabsf(loud[BATCH * NSEG + idx] - loud[idx]);
    sm[threadIdx.x] = s;
    __syncthreads();
    for (int off = 128; off > 0; off >>= 1) {
        if (threadIdx.x < off) sm[threadIdx.x] += sm[threadIdx.x + off];
        __syncthreads();
    }
    if (threadIdx.x == 0) band_loss[band] = sm[0] / (float)(BATCH * NSEG);
}

__global__ void tfl_final_kernel(const float* __restrict__ band_loss, float* __restrict__ out) {
    if (threadIdx.x == 0) {
        float s = 0.0f;
        for (int b = 0; b < NBANDS; ++b) s += band_loss[b];
        out[0] = s / (float)NBANDS;
    }
}

// ---------------------------------------------------------------------------
// Host side
// ---------------------------------------------------------------------------
static void hp_coeffs_host(double cutoff, double* c) {
    const double Q = 0.707, SR = 16000.0;
    double w0 = 2.0 * M_PI * cutoff / SR;
    double alpha = sin(w0) / (2.0 * Q);
    double cw = cos(w0);
    double b0 = (1.0 + cw) / 2.0, b1 = -(1.0 + cw), b2 = (1.0 + cw) / 2.0;
    double a0 = 1.0 + alpha, a1 = -2.0 * cw, a2 = 1.0 - alpha;
    c[0] = (double)(float)(b0 / a0); c[1] = (double)(float)(b1 / a0);
    c[2] = (double)(float)(b2 / a0); c[3] = (double)(float)(a1 / a0);
    c[4] = (double)(float)(a2 / a0);
}
static void lp_coeffs_host(double cutoff, double* c) {
    const double Q = 0.707, SR = 16000.0;
    double w0 = 2.0 * M_PI * cutoff / SR;
    double alpha = sin(w0) / (2.0 * Q);
    double cw = cos(w0);
    double b0 = (1.0 - cw) / 2.0, b1 = 1.0 - cw, b2 = (1.0 - cw) / 2.0;
    double a0 = 1.0 + alpha, a1 = -2.0 * cw, a2 = 1.0 - alpha;
    c[0] = (double)(float)(b0 / a0); c[1] = (double)(float)(b1 / a0);
    c[2] = (double)(float)(b2 / a0); c[3] = (double)(float)(a1 / a0);
    c[4] = (double)(float)(a2 / a0);
}

extern "C" void kernel_launch(void* const* d_in, const int* in_sizes, int n_in,
                              void* d_out, int out_size, void* d_ws, size_t ws_size,
                              hipStream_t stream) {
    (void)in_sizes; (void)n_in; (void)out_size; (void)ws_size;
    const float* sig = (const float*)d_in[0];
    const float* wm  = (const float*)d_in[1];
    float* out = (float*)d_out;

    // workspace carve-up (256B aligned)
    uint8_t* w = (uint8_t*)d_ws;
    size_t off = 0;
    auto take = [&](size_t bytes) -> void* {
        void* p = (void*)(w + off);
        off = (off + bytes + 255) & ~(size_t)255;
        return p;
    };
    float*  hws       = (float*)take(KTAPS * sizeof(float));
    double* Mws       = (double*)take(NBANDS * 36 * sizeof(double));
    double* Szs       = (double*)take((size_t)2 * BATCH * NCHUNK * 6 * sizeof(double));
    float*  Sstart    = (float*)take((size_t)2 * BATCH * NCHUNK * 6 * sizeof(float));
    float*  yband     = (float*)take((size_t)2 * BATCH * T_USE * sizeof(float));
    float*  loud      = (float*)take((size_t)2 * BATCH * NSEG * sizeof(float));
    float*  band_loss = (float*)take(NBANDS * sizeof(float));

    // per-band coefficients, rounded through f32 like the reference
    AllC allc;
    C10d cfd[NBANDS];
    C10f cff[NBANDS];
    for (int b = 0; b < NBANDS; ++b) {
        double low  = b * 16000.0 / (2.0 * NBANDS);
        double high = (b + 1) * 16000.0 / (2.0 * NBANDS);
        hp_coeffs_host(low,  &cfd[b].v[0]);
        lp_coeffs_host(high, &cfd[b].v[5]);
        for (int k = 0; k < 10; ++k) {
            allc.v[b][k] = cfd[b].v[k];
            cff[b].v[k]  = (float)cfd[b].v[k];
        }
    }

    tfl_init_kernel<<<1, 64, 0, stream>>>(allc, Mws, hws);

    const int nthread = 2 * BATCH * NCHUNK;           // 9984
    const int nblk    = (nthread + 255) / 256;        // 39
    for (int band = 0; band < NBANDS; ++band) {
        tfl_pass1_kernel<<<nblk, 256, 0, stream>>>(sig, wm, cfd[band], Szs);
        tfl_scan_kernel<<<1, 64, 0, stream>>>(Mws + band * 36, Szs, Sstart);
        tfl_pass2_kernel<<<nblk, 256, 0, stream>>>(sig, wm, cff[band], Sstart, yband);
        tfl_wmma_loud_kernel<<<2 * BATCH * SEG_GROUPS, 32, 0, stream>>>(yband, hws, loud);
        tfl_band_reduce_kernel<<<1, 256, 0, stream>>>(loud, band_loss, band);
    }
    tfl_final_kernel<<<1, 32, 0, stream>>>(band_loss, out);
}